// LowRankDiagLightSBPotential_44581760533093
// MI455X (gfx1250) — compile-verified
//
#include <hip/hip_runtime.h>
#include <math.h>

typedef float v2f __attribute__((ext_vector_type(2)));
typedef float v8f __attribute__((ext_vector_type(8)));

#define K_COMP   64
#define D_DIM    128
#define R_RANK   8
#define NCOLS    640     // 64 (q0) + 64 (q1) + 512 (t)
#define P_STRIDE 80      // 64 Minv + 8 c + 1 A2 (padded)
#define LOG2PI   1.8378770664093453f

// ---------------------------------------------------------------------------
// Prep kernel: per-component precompute. Thread k handles Gaussian k.
// Produces:
//   W2 [NCOLS][D_DIM] : fused GEMM weight matrix (column-major-by-n rows)
//       col k        : -0.5 * Sinv_{k,d}            (multiplied by y^2)
//       col 64+k     : Sinv_{k,d} * m_{k,d}         (multiplied by y)
//       col 128+8k+r : V_{k,d,r} = Sinv_{k,d}*U      (multiplied by y)
//   P  [K][P_STRIDE]  : Minv (8x8), c_r = V^T m (8), A2 (folded constant)
// ---------------------------------------------------------------------------
__global__ void lrdl_prep_kernel(const float* __restrict__ m,
                                 const float* __restrict__ delta,
                                 const float* __restrict__ U,
                                 const float* __restrict__ lar,
                                 float* __restrict__ W2,
                                 float* __restrict__ P)
{
    const int k = threadIdx.x;
    if (k >= K_COMP) return;

    float Sinv[D_DIM];
    float sumdelta = 0.f;
    for (int d = 0; d < D_DIM; ++d) {
        float dl = delta[k * D_DIM + d];
        sumdelta += dl;
        Sinv[d] = expf(-dl);
    }

    // Capacitance M = I + U^T diag(Sinv) U ; also c = V^T m, mq = m^T Sinv m
    float M[R_RANK][R_RANK];
    float c[R_RANK];
    float mq = 0.f;
    for (int r = 0; r < R_RANK; ++r) {
        c[r] = 0.f;
        for (int s = 0; s < R_RANK; ++s) M[r][s] = (r == s) ? 1.f : 0.f;
    }
    for (int d = 0; d < D_DIM; ++d) {
        float u[R_RANK];
        const float si = Sinv[d];
        const float md = m[k * D_DIM + d];
        for (int r = 0; r < R_RANK; ++r) u[r] = U[(k * D_DIM + d) * R_RANK + r];
        for (int r = 0; r < R_RANK; ++r) {
            const float ur_si = u[r] * si;
            c[r] += ur_si * md;
            for (int s = 0; s < R_RANK; ++s) M[r][s] += ur_si * u[s];
        }
        mq += md * md * si;
    }

    // Cholesky M = L L^T
    float L[R_RANK][R_RANK];
    float sumlogL = 0.f;
    for (int i = 0; i < R_RANK; ++i) {
        for (int j = 0; j <= i; ++j) {
            float s = M[i][j];
            for (int p = 0; p < j; ++p) s -= L[i][p] * L[j][p];
            if (j == i) {
                L[i][i] = sqrtf(s);
                sumlogL += logf(L[i][i]);
            } else {
                L[i][j] = s / L[j][j];
            }
        }
    }
    const float logdetS = sumdelta + 2.f * sumlogL;

    // Linv (lower-triangular inverse), then Minv = Linv^T Linv
    float Li[R_RANK][R_RANK];
    for (int i = 0; i < R_RANK; ++i) {
        Li[i][i] = 1.f / L[i][i];
        for (int j = 0; j < i; ++j) {
            float s = 0.f;
            for (int p = j; p < i; ++p) s += L[i][p] * Li[p][j];
            Li[i][j] = -s / L[i][i];
        }
    }
    float Minv[R_RANK][R_RANK];
    for (int r = 0; r < R_RANK; ++r)
        for (int s = 0; s < R_RANK; ++s) {
            float acc = 0.f;
            int p0 = (r > s) ? r : s;
            for (int p = p0; p < R_RANK; ++p) acc += Li[p][r] * Li[p][s];
            Minv[r][s] = acc;
        }

    // log_alpha = lar - mean(lar)  (EPSILON == 1)
    float mean = 0.f;
    for (int i = 0; i < K_COMP; ++i) mean += lar[i];
    mean *= (1.f / K_COMP);
    const float la = lar[k] - mean;

    const float A2 = la - 0.5f * ((float)D_DIM * LOG2PI + logdetS) - 0.5f * mq;

    // Emit fused weight matrix
    for (int d = 0; d < D_DIM; ++d) {
        const float si = Sinv[d];
        W2[(size_t)k * D_DIM + d]            = -0.5f * si;
        W2[(size_t)(64 + k) * D_DIM + d]     = si * m[k * D_DIM + d];
        for (int r = 0; r < R_RANK; ++r)
            W2[(size_t)(128 + k * R_RANK + r) * D_DIM + d] =
                si * U[(k * D_DIM + d) * R_RANK + r];
    }
    float* Pk = P + k * P_STRIDE;
    for (int r = 0; r < R_RANK; ++r)
        for (int s = 0; s < R_RANK; ++s) Pk[r * R_RANK + s] = Minv[r][s];
    for (int r = 0; r < R_RANK; ++r) Pk[64 + r] = c[r];
    Pk[72] = A2;
}

// ---------------------------------------------------------------------------
// Main kernel: one block = 16 rows of y. 8 waves; wave w owns N-tiles
// {w, w+8, w+16, w+24, w+32} of the 40 16-col tiles (640 cols total).
// GEMM via V_WMMA_F32_16X16X4_F32, D=128 -> 32 K-steps.
// ---------------------------------------------------------------------------
#define YSTRIDE 132   // 128 + 4 pad: kills LDS bank conflicts on A reads
#define SSTRIDE 644   // 640 + 4 pad: half-waves hit disjoint bank sets

__global__ __launch_bounds__(256)
void lrdl_logz_kernel(const float* __restrict__ y,
                      const float* __restrict__ W2,
                      const float* __restrict__ P,
                      float* __restrict__ out)
{
    __shared__ float ldsY[16 * YSTRIDE];
    __shared__ float ldsS[16 * SSTRIDE];
    __shared__ float ldsL[16 * 64];

    const int tid  = threadIdx.x;
    const int b0   = blockIdx.x * 16;

    // Stage 16x128 y tile into padded LDS (global reads fully coalesced).
    for (int e = tid; e < 16 * D_DIM; e += 256) {
        int row = e >> 7, d = e & 127;
        ldsY[row * YSTRIDE + d] = y[(size_t)b0 * D_DIM + e];
    }
    __syncthreads();

    const int lane  = tid & 31;
    const int w     = tid >> 5;
    const int mrow  = lane & 15;
    const int khalf = (lane >> 4) << 1;      // 0 or 2 (K sub-offset per ISA A layout)

    v8f acc[5] = {};

    // Warm the L2 path for our B-fragment columns.
    __builtin_prefetch(W2 + (size_t)(w * 16) * D_DIM, 0, 1);

    for (int kk = 0; kk < 32; ++kk) {
        const int dA = kk * 4 + khalf;
        const float a0 = ldsY[mrow * YSTRIDE + dA];
        const float a1 = ldsY[mrow * YSTRIDE + dA + 1];
        v2f ay;  ay.x  = a0;      ay.y  = a1;
        v2f ay2; ay2.x = a0 * a0; ay2.y = a1 * a1;

#pragma unroll
        for (int i = 0; i < 5; ++i) {
            const int tn  = w + 8 * i;                 // N-tile index 0..39
            const int col = tn * 16 + mrow;            // column 0..639
            const v2f b = *(const v2f*)(W2 + (size_t)col * D_DIM + dA);
            const v2f a = (tn < 4) ? ay2 : ay;         // q0 tiles consume y^2
            acc[i] = __builtin_amdgcn_wmma_f32_16x16x4_f32(
                false, a, false, b, (short)0, acc[i], false, false);
        }
    }

    // Scatter C fragments to staging LDS (C layout: VGPR v -> M = v + 8*(lane/16))
#pragma unroll
    for (int i = 0; i < 5; ++i) {
        const int col = (w + 8 * i) * 16 + mrow;
#pragma unroll
        for (int v = 0; v < 8; ++v) {
            const int row = v + ((lane >> 4) << 3);
            ldsS[row * SSTRIDE + col] = acc[i][v];
        }
    }
    __syncthreads();

    // Epilogue: logits for 16 rows x 64 components (4 pairs per thread)
    for (int p = tid; p < 16 * 64; p += 256) {
        const int row = p >> 6;
        const int k   = p & 63;
        const float* __restrict__ Pk = P + k * P_STRIDE;

        float t[R_RANK];
#pragma unroll
        for (int r = 0; r < R_RANK; ++r)
            t[r] = ldsS[row * SSTRIDE + 128 + k * R_RANK + r] - Pk[64 + r];

        float g = 0.f;
#pragma unroll
        for (int r = 0; r < R_RANK; ++r) {
            float s = 0.f;
#pragma unroll
            for (int q = 0; q < R_RANK; ++q) s += Pk[r * R_RANK + q] * t[q];
            g += t[r] * s;
        }
        // logit = A2 + (-0.5*q0) + q1 + 0.5 * t^T Minv t
        const float logit = Pk[72] + ldsS[row * SSTRIDE + k]
                          + ldsS[row * SSTRIDE + 64 + k] + 0.5f * g;
        ldsL[row * 64 + k] = logit;
    }
    __syncthreads();

    // Per-row logsumexp over K=64
    if (tid < 16) {
        float mx = -INFINITY;
        for (int k = 0; k < 64; ++k) mx = fmaxf(mx, ldsL[tid * 64 + k]);
        float sum = 0.f;
        for (int k = 0; k < 64; ++k) sum += expf(ldsL[tid * 64 + k] - mx);
        out[b0 + tid] = mx + logf(sum);
    }
}

// ---------------------------------------------------------------------------
extern "C" void kernel_launch(void* const* d_in, const int* in_sizes, int n_in,
                              void* d_out, int out_size, void* d_ws, size_t ws_size,
                              hipStream_t stream) {
    const float* y     = (const float*)d_in[0];
    const float* m     = (const float*)d_in[1];
    const float* delta = (const float*)d_in[2];
    const float* U     = (const float*)d_in[3];
    const float* lar   = (const float*)d_in[4];
    float* out = (float*)d_out;

    float* W2 = (float*)d_ws;                               // 640*128 floats
    float* P  = W2 + (size_t)NCOLS * D_DIM;                 // 64*80 floats

    const int nB = in_sizes[0] / D_DIM;                     // 8192
    lrdl_prep_kernel<<<1, 64, 0, stream>>>(m, delta, U, lar, W2, P);
    lrdl_logz_kernel<<<nB / 16, 256, 0, stream>>>(y, W2, P, out);
}